// My_GNN_GNN_NN_28647431864458
// MI455X (gfx1250) — compile-verified
//
#include <hip/hip_runtime.h>

#define NODES 100000
#define EDGES 1600000

typedef __attribute__((ext_vector_type(2))) float v2f;
typedef __attribute__((ext_vector_type(8))) float v8f;

// ---------------------------------------------------------------- utilities

__global__ void zero_f32(float* __restrict__ p, int n) {
    int i = blockIdx.x * blockDim.x + threadIdx.x;
    if (i < n) p[i] = 0.f;
}

__global__ void deg_accum(const int* __restrict__ col, float* __restrict__ deg, int e) {
    int i = blockIdx.x * blockDim.x + threadIdx.x;
    if (i < e) atomicAdd(&deg[col[i]], 1.0f);
}

__global__ void deg_to_dis(float* __restrict__ deg, int n) {
    int i = blockIdx.x * blockDim.x + threadIdx.x;
    if (i < n) {
        float d = deg[i];
        deg[i] = (d > 0.f) ? rsqrtf(d) : 0.f;
    }
}

__global__ void edge_norm(const int* __restrict__ row, const int* __restrict__ col,
                          const float* __restrict__ dis, float* __restrict__ nrm, int e) {
    int i = blockIdx.x * blockDim.x + threadIdx.x;
    if (i < e) nrm[i] = dis[row[i]] * dis[col[i]];
}

// ------------------------------------------------- WMMA f32 16x16x4 GEMM
// C[M x fout] = act(A[M x fin] @ W + bias)
// One wave (32 lanes) computes one 16x16 output tile.
// packed==1: W is [K, fin, 32] (ARMA stacks, fout = K*32, 16-col tile sits
// inside one stack); packed==0: W is [fin, fout] row-major.
// M (==NODES) and fout are multiples of 16; fin is a multiple of 4.
__global__ __launch_bounds__(32) void gemm16_wmma(
    const float* __restrict__ A, int lda,
    const float* __restrict__ W, const float* __restrict__ bias,
    float* __restrict__ C, int ldc,
    int fin, int fout, int packed, int do_relu) {

    const int lane = threadIdx.x;      // 0..31
    const int cl   = lane & 15;
    const int hi   = lane >> 4;        // 0 or 1
    const int c0   = blockIdx.x * 16;
    const int m0   = blockIdx.y * 16;
    const int colLane = c0 + cl;

    const float* Wp;
    int wstride;
    if (packed) {
        int k = c0 >> 5;                               // stack index (uniform per tile)
        Wp = W + (size_t)k * fin * 32 + (colLane & 31);
        wstride = 32;
    } else {
        Wp = W + colLane;
        wstride = fout;
    }

    // C/D layout: VGPR j -> (lanes 0-15: M=j, N=lane) (lanes 16-31: M=8+j, N=lane-16)
    v8f acc;
    float bv = bias ? bias[colLane] : 0.f;
#pragma unroll
    for (int j = 0; j < 8; ++j) acc[j] = bv;

    const float* Ap = A + (size_t)(m0 + cl) * lda;
    const int khalf = hi << 1;                          // 0 or 2

    for (int kk = 0; kk < fin; kk += 4) {
        const int k0 = kk + khalf;
        // A 16x4 frag: lanes 0-15 hold K=0/1, lanes 16-31 hold K=2/3
        v2f a;
        a[0] = Ap[k0];
        a[1] = Ap[k0 + 1];
        // B 4x16 frag: mirrored striping across lanes
        v2f b;
        b[0] = Wp[(size_t)k0 * wstride];
        b[1] = Wp[(size_t)(k0 + 1) * wstride];
        acc = __builtin_amdgcn_wmma_f32_16x16x4_f32(false, a, false, b,
                                                    (short)0, acc, false, false);
    }

    float* Cp = C + (size_t)(m0 + (hi << 3)) * ldc + colLane;
#pragma unroll
    for (int j = 0; j < 8; ++j) {
        float v = acc[j];
        if (do_relu) v = fmaxf(v, 0.f);
        Cp[(size_t)j * ldc] = v;
    }
}

// --------------------------------------------- ARMA edge scatter-accumulate
// agg[col[e], :] += norm[e] * hk[row[e], :]   (128 channels, L2-resident)
// thread = (edge, group of 4 channels)
__global__ void scatter_edges(const int* __restrict__ row, const int* __restrict__ col,
                              const float* __restrict__ nrm,
                              const float* __restrict__ hk, float* __restrict__ agg, int e) {
    long idx = (long)blockIdx.x * blockDim.x + threadIdx.x;
    if (idx >= (long)e * 32) return;
    int ed = (int)(idx >> 5);
    int g  = (int)(idx & 31);
    int r = row[ed];
    int c = col[ed];
    float nv = nrm[ed];
    const float4 h = *(const float4*)(hk + (size_t)r * 128 + g * 4);
    float* dst = agg + (size_t)c * 128 + g * 4;
    atomicAdd(dst + 0, nv * h.x);
    atomicAdd(dst + 1, nv * h.y);
    atomicAdd(dst + 2, nv * h.z);
    atomicAdd(dst + 3, nv * h.w);
}

// relu per stack, mean over K=4 stacks, outer relu
__global__ void arma_finish(const float* __restrict__ agg, float* __restrict__ out32, int n) {
    int idx = blockIdx.x * blockDim.x + threadIdx.x;
    if (idx >= n * 32) return;
    int nn = idx >> 5;
    int o  = idx & 31;
    const float* a = agg + (size_t)nn * 128 + o;
    float s = fmaxf(a[0], 0.f) + fmaxf(a[32], 0.f) + fmaxf(a[64], 0.f) + fmaxf(a[96], 0.f);
    out32[idx] = fmaxf(0.25f * s, 0.f);
}

// ----------------------------------------------------- fused fc3+fc4 tail
__global__ void mlp_tail(const float* __restrict__ h32,
                         const float* __restrict__ w3, const float* __restrict__ b3,
                         const float* __restrict__ w4, const float* __restrict__ b4,
                         float* __restrict__ out, int n) {
    int i = blockIdx.x * blockDim.x + threadIdx.x;
    if (i >= n) return;
    const float* h = h32 + (size_t)i * 32;
    float t[16];
#pragma unroll
    for (int o = 0; o < 16; ++o) {
        float s = b3[o];
#pragma unroll
        for (int f = 0; f < 32; ++f) s += h[f] * w3[f * 16 + o];
        t[o] = fmaxf(s, 0.f);
    }
    float o0 = b4[0], o1 = b4[1];
#pragma unroll
    for (int f = 0; f < 16; ++f) {
        o0 += t[f] * w4[f * 2 + 0];
        o1 += t[f] * w4[f * 2 + 1];
    }
    out[(size_t)i * 2 + 0] = o0;
    out[(size_t)i * 2 + 1] = o1;
}

// ---------------------------------------------------------------- host side

static inline int cdiv(long a, long b) { return (int)((a + b - 1) / b); }

static void arma_layer(const float* X, int fin,
                       const float* iw, const float* rw, const float* bb,
                       const int* row, const int* col, const float* nrm,
                       float* hk, float* agg, float* out32, hipStream_t stream) {
    dim3 g(128 / 16, NODES / 16);
    // hk = X @ iw   (packed [K,fin,32], no bias, no relu)
    gemm16_wmma<<<g, 32, 0, stream>>>(X, fin, iw, nullptr, hk, 128, fin, 128, 1, 0);
    // agg = X @ rw + b  (pre-initializes the segment-sum accumulator)
    gemm16_wmma<<<g, 32, 0, stream>>>(X, fin, rw, bb, agg, 128, fin, 128, 1, 0);
    long t = (long)EDGES * 32;
    scatter_edges<<<cdiv(t, 256), 256, 0, stream>>>(row, col, nrm, hk, agg, EDGES);
    arma_finish<<<cdiv((long)NODES * 32, 256), 256, 0, stream>>>(agg, out32, NODES);
}

extern "C" void kernel_launch(void* const* d_in, const int* in_sizes, int n_in,
                              void* d_out, int out_size, void* d_ws, size_t ws_size,
                              hipStream_t stream) {
    const float* x     = (const float*)d_in[0];
    const int*   ei    = (const int*)d_in[1];
    const int*   row   = ei;
    const int*   col   = ei + EDGES;
    const float* fc1_w = (const float*)d_in[2];
    const float* fc1_b = (const float*)d_in[3];
    const float* fc2_w = (const float*)d_in[4];
    const float* fc2_b = (const float*)d_in[5];
    const float* a1_iw = (const float*)d_in[6];
    const float* a1_rw = (const float*)d_in[7];
    const float* a1_b  = (const float*)d_in[8];
    const float* a2_iw = (const float*)d_in[9];
    const float* a2_rw = (const float*)d_in[10];
    const float* a2_b  = (const float*)d_in[11];
    const float* a3_iw = (const float*)d_in[12];
    const float* a3_rw = (const float*)d_in[13];
    const float* a3_b  = (const float*)d_in[14];
    const float* a4_iw = (const float*)d_in[15];
    const float* a4_rw = (const float*)d_in[16];
    const float* a4_b  = (const float*)d_in[17];
    const float* fc3_w = (const float*)d_in[18];
    const float* fc3_b = (const float*)d_in[19];
    const float* fc4_w = (const float*)d_in[20];
    const float* fc4_b = (const float*)d_in[21];

    float* ws   = (float*)d_ws;
    float* h128 = ws;                                  // N*128
    float* hk   = ws + (size_t)NODES * 128;            // N*128
    float* agg  = ws + (size_t)NODES * 128 * 2;        // N*128
    float* h32a = ws + (size_t)NODES * 128 * 3;        // N*32
    float* h32b = h32a + (size_t)NODES * 32;           // N*32
    float* deg  = h32b + (size_t)NODES * 32;           // N
    float* nrm  = deg + NODES;                         // E

    // gcn_norm (no self loops)
    zero_f32 <<<cdiv(NODES, 256), 256, 0, stream>>>(deg, NODES);
    deg_accum<<<cdiv(EDGES, 256), 256, 0, stream>>>(col, deg, EDGES);
    deg_to_dis<<<cdiv(NODES, 256), 256, 0, stream>>>(deg, NODES);
    edge_norm<<<cdiv(EDGES, 256), 256, 0, stream>>>(row, col, deg, nrm, EDGES);

    // fc1: [N,4] @ [4,32] -> h32a (relu)
    gemm16_wmma<<<dim3(32 / 16, NODES / 16), 32, 0, stream>>>(
        x, 4, fc1_w, fc1_b, h32a, 32, 4, 32, 0, 1);
    // fc2: [N,32] @ [32,128] -> h128 (relu)
    gemm16_wmma<<<dim3(128 / 16, NODES / 16), 32, 0, stream>>>(
        h32a, 32, fc2_w, fc2_b, h128, 128, 32, 128, 0, 1);

    // 4 ARMA layers
    arma_layer(h128, 128, a1_iw, a1_rw, a1_b, row, col, nrm, hk, agg, h32a, stream);
    arma_layer(h32a,  32, a2_iw, a2_rw, a2_b, row, col, nrm, hk, agg, h32b, stream);
    arma_layer(h32b,  32, a3_iw, a3_rw, a3_b, row, col, nrm, hk, agg, h32a, stream);
    arma_layer(h32a,  32, a4_iw, a4_rw, a4_b, row, col, nrm, hk, agg, h32b, stream);

    // fc3 + fc4 tail -> d_out [N,2]
    mlp_tail<<<cdiv(NODES, 256), 256, 0, stream>>>(
        h32b, fc3_w, fc3_b, fc4_w, fc4_b, (float*)d_out, NODES);
}